// TopkGatherRsTritonOp_44848048505530
// MI455X (gfx1250) — compile-verified
//
#include <hip/hip_runtime.h>
#include <cstdint>
#include <cstddef>

// Problem constants (match the reference file)
constexpr int kTokens  = 8192;
constexpr int kTopK    = 2;
constexpr int kHidden  = 4096;
constexpr int kMFull   = kTokens * kTopK;   // 16384

// Kernel tiling
constexpr int kBlock   = 256;               // 8 wave32
constexpr int kChunk   = kBlock * 4;        // 1024 floats per chunk (one b128 per lane per row)
constexpr int kNChunks = kHidden / kChunk;  // 4

typedef float v4f __attribute__((ext_vector_type(4)));

// out[t,h] = sum over 4 gathered rows: (g0,i0),(g0,i1),(g1,i0),(g1,i1)
// Data path: HBM --(global_load_async_to_lds_b128, ASYNCcnt)--> LDS --ds_load_b128--> VGPR
//            3x v_add_f32 per element --> global_store_b128 (non-temporal)
// Each lane async-loads exactly the 16B slices it later consumes, so no
// workgroup barrier is needed; per-wave s_wait_asynccnt provides ordering
// (async loads complete in order).
__global__ __launch_bounds__(kBlock)
void topk_gather_rs_kernel(const float* __restrict__ gbuf,
                           const long long* __restrict__ sidx,
                           float* __restrict__ out)
{
    // [chunk][row][floats] : 4 * 4 * 1024 * 4B = 64 KB (fits; 320 KB LDS/WGP)
    __shared__ __align__(16) float smem[kNChunks][4][kChunk];

    const uint32_t t   = blockIdx.x;
    const uint32_t tid = threadIdx.x;

    // Uniform index loads (int64) -> scalar loads
    const long long i0 = sidx[2 * (size_t)t + 0];
    const long long i1 = sidx[2 * (size_t)t + 1];

    // gemm_buffer layout: (2, M_FULL, HIDDEN) row-major f32
    const float* r0 = gbuf + (size_t)i0 * kHidden;                       // group 0, topk 0
    const float* r1 = gbuf + (size_t)i1 * kHidden;                       // group 0, topk 1
    const float* r2 = gbuf + ((size_t)kMFull + (size_t)i0) * kHidden;    // group 1, topk 0
    const float* r3 = gbuf + ((size_t)kMFull + (size_t)i1) * kHidden;    // group 1, topk 1

    // Low 32 bits of a generic pointer to LDS == LDS byte address (aperture layout)
    const uint32_t ldsBase  = (uint32_t)(uintptr_t)(&smem[0][0][0]);
    const uint32_t laneByte = tid * 16u;                 // this lane's 16B slot within a row
    constexpr uint32_t kRowBytes   = kChunk * 4u;        // 4 KB per row slice
    constexpr uint32_t kChunkBytes = 4u * kRowBytes;     // 16 KB per chunk (4 rows)

    // ---- Issue ALL async loads up front: 16 per lane, 64 KB per block in flight,
    // ---- zero VGPRs held for in-flight data (tracked by ASYNCcnt).
#pragma unroll
    for (int c = 0; c < kNChunks; ++c) {
        const uint32_t col = (uint32_t)c * kChunk + tid * 4u;   // float column this lane covers
        const uint32_t lb  = ldsBase + (uint32_t)c * kChunkBytes + laneByte;
        asm volatile("global_load_async_to_lds_b128 %0, %1, off"
                     :: "v"(lb + 0u * kRowBytes),
                        "v"((uint64_t)(uintptr_t)(r0 + col)) : "memory");
        asm volatile("global_load_async_to_lds_b128 %0, %1, off"
                     :: "v"(lb + 1u * kRowBytes),
                        "v"((uint64_t)(uintptr_t)(r1 + col)) : "memory");
        asm volatile("global_load_async_to_lds_b128 %0, %1, off"
                     :: "v"(lb + 2u * kRowBytes),
                        "v"((uint64_t)(uintptr_t)(r2 + col)) : "memory");
        asm volatile("global_load_async_to_lds_b128 %0, %1, off"
                     :: "v"(lb + 3u * kRowBytes),
                        "v"((uint64_t)(uintptr_t)(r3 + col)) : "memory");
    }

    v4f* outp = (v4f*)(out + (size_t)t * kHidden);
    constexpr int kV4PerRow = kChunk / 4;                // 256 v4f per row slice

    // ---- Consume with a sliding wait: before chunk c, only the (kNChunks-1-c)
    // ---- later chunks' 4 loads each may still be outstanding (in-order retire).
#pragma unroll
    for (int c = 0; c < kNChunks; ++c) {
        asm volatile("s_wait_asynccnt %0" :: "i"((kNChunks - 1 - c) * 4) : "memory");
        const v4f* s = (const v4f*)(&smem[c][0][0]);
        v4f a = s[0 * kV4PerRow + tid];
        v4f b = s[1 * kV4PerRow + tid];
        v4f d = s[2 * kV4PerRow + tid];
        v4f e = s[3 * kV4PerRow + tid];
        v4f r = (a + b) + (d + e);
        // Output is written once and never re-read: non-temporal store so the
        // 128 MB result doesn't thrash the 192 MB L2 against the 512 MB read stream.
        __builtin_nontemporal_store(r, &outp[(size_t)c * kV4PerRow + tid]);
    }
}

extern "C" void kernel_launch(void* const* d_in, const int* in_sizes, int n_in,
                              void* d_out, int out_size, void* d_ws, size_t ws_size,
                              hipStream_t stream)
{
    const float*     gbuf = (const float*)d_in[0];       // (2, 16384, 4096) f32
    const long long* sidx = (const long long*)d_in[1];   // (8192, 2) i64
    float*           out  = (float*)d_out;               // (8192, 4096) f32

    topk_gather_rs_kernel<<<kTokens, kBlock, 0, stream>>>(gbuf, sidx, out);
}